// resnext_layers_c_6339371729592
// MI455X (gfx1250) — compile-verified
//
#include <hip/hip_runtime.h>

#define EPSV 1e-3f

typedef float v2f __attribute__((ext_vector_type(2)));
typedef float v4f __attribute__((ext_vector_type(4)));
typedef float v8f __attribute__((ext_vector_type(8)));

// Problem dims (fixed by the reference)
#define BATCH 32
#define HWDIM 56
#define C0 256
#define C1 128
#define C2 256
#define TILE 14       // output tile edge (56 = 4*14)
#define HALO 16       // TILE + 2 (3x3 SAME halo)
#define NPIX_H 256    // HALO*HALO
#define NPIX_T 196    // TILE*TILE
#define MPAD3 208     // ceil(196/16)*16

// LDS layout (float offsets)
#define OFF_H1 0
#define OFF_H2 (OFF_H1 + NPIX_H * C1)   // 32768
#define OFF_WG (OFF_H2 + MPAD3 * C1)    // 59392
#define OFF_S1 (OFF_WG + 9 * 4 * C1)    // 64000
#define OFF_T1 (OFF_S1 + C1)
#define OFF_S2 (OFF_T1 + C1)
#define OFF_T2 (OFF_S2 + C1)
#define OFF_S3 (OFF_T2 + C1)
#define OFF_T3 (OFF_S3 + C2)
#define LDS_FLOATS (OFF_T3 + C2)        // 65024 floats = 260096 B < 320 KB
#define LDS_BYTES (LDS_FLOATS * 4)

__launch_bounds__(256, 1)
__global__ void resnext_fused_kernel(
    const float* __restrict__ x,
    const float* __restrict__ w1, const float* __restrict__ b1,
    const float* __restrict__ g1, const float* __restrict__ be1,
    const float* __restrict__ m1, const float* __restrict__ v1,
    const float* __restrict__ wg, const float* __restrict__ bg,
    const float* __restrict__ g2, const float* __restrict__ be2,
    const float* __restrict__ m2, const float* __restrict__ v2,
    const float* __restrict__ w3, const float* __restrict__ b3,
    const float* __restrict__ g3, const float* __restrict__ be3,
    const float* __restrict__ m3, const float* __restrict__ v3,
    float* __restrict__ out)
{
    extern __shared__ float lds[];

    const int tid   = threadIdx.x;
    const int lane  = tid & 31;        // wave32
    const int wave  = tid >> 5;        // 8 waves / block
    const int lhalf = lane >> 4;       // 0 or 1 (K-pair half)
    const int l16   = lane & 15;

    const int blk = blockIdx.x;        // 512 blocks: 32 images x 16 tiles
    const int b   = blk >> 4;
    const int t   = blk & 15;
    const int ty0 = (t >> 2) * TILE;
    const int tx0 = (t & 3) * TILE;

    // ---- Phase 0: fold bias+BN into scale/shift; stage grouped weights ----
    {
        const int c = tid;
        if (c < C2) {
            const float inv3 = g3[c] * rsqrtf(v3[c] + EPSV);
            lds[OFF_S3 + c] = inv3;
            lds[OFF_T3 + c] = be3[c] + (b3[c] - m3[c]) * inv3;
        }
        if (c < C1) {
            const float inv1 = g1[c] * rsqrtf(v1[c] + EPSV);
            lds[OFF_S1 + c] = inv1;
            lds[OFF_T1 + c] = be1[c] + (b1[c] - m1[c]) * inv1;
            const float inv2 = g2[c] * rsqrtf(v2[c] + EPSV);
            lds[OFF_S2 + c] = inv2;
            lds[OFF_T2 + c] = be2[c] + (bg[c] - m2[c]) * inv2;
        }
        for (int i = tid; i < 9 * 4 * C1; i += 256)
            lds[OFF_WG + i] = wg[i];
        // zero pad rows 196..207 of h2 (keeps conv3 A-frag reads defined)
        for (int i = tid; i < (MPAD3 - NPIX_T) * C1; i += 256)
            lds[OFF_H2 + NPIX_T * C1 + i] = 0.0f;
    }
    __syncthreads();

    // ---- Phase 1: conv1x1 (256->128) via WMMA; M = 256 halo pixels -------
    // M-tile mt == halo row; A row m == halo column.
    for (int tt = wave; tt < 16 * 8; tt += 8) {
        const int mt = tt >> 3;               // halo row 0..15
        const int nt = tt & 7;                // channel tile 0..7
        const int gy = ty0 - 1 + mt;
        const int gx = tx0 - 1 + l16;
        const bool valid = (gy >= 0) && (gy < HWDIM) && (gx >= 0) && (gx < HWDIM);
        const int gyc = min(max(gy, 0), HWDIM - 1);
        const int gxc = min(max(gx, 0), HWDIM - 1);
        const float* xrow = x + (((size_t)b * HWDIM + gyc) * HWDIM + gxc) * C0;
        const int n = nt * 16 + l16;

        v8f acc = {0.f, 0.f, 0.f, 0.f, 0.f, 0.f, 0.f, 0.f};
        #pragma unroll 8
        for (int ks = 0; ks < C0 / 4; ++ks) {
            const int k = ks * 4 + lhalf * 2;
            v2f a = *(const v2f*)(xrow + k);          // global_load_b64
            if (!valid) { a.x = 0.f; a.y = 0.f; }     // SAME zero padding (cndmask, EXEC all 1s)
            v2f bb;
            bb.x = w1[k * C1 + n];
            bb.y = w1[(k + 1) * C1 + n];
            acc = __builtin_amdgcn_wmma_f32_16x16x4_f32(
                false, a, false, bb, (short)0, acc, false, false);
        }
        const float sc = lds[OFF_S1 + n];
        const float sh = lds[OFF_T1 + n];
        #pragma unroll
        for (int r = 0; r < 8; ++r) {
            const int m = r + lhalf * 8;              // halo column
            float vv = acc[r] * sc + sh;
            vv = vv > 0.f ? vv : 0.f;
            lds[OFF_H1 + (mt * 16 + m) * C1 + n] = vv;
        }
    }
    __syncthreads();

    // ---- Phase 2: grouped 3x3 conv (32 groups x 4ch), VALU from LDS ------
    {
        const int c  = tid & 127;                     // fixed channel per thread
        const int g4 = (c >> 2) * 4;                  // group base channel
        float wreg[36];
        #pragma unroll
        for (int j = 0; j < 9; ++j)
            #pragma unroll
            for (int i = 0; i < 4; ++i)
                wreg[j * 4 + i] = lds[OFF_WG + j * 512 + i * 128 + c];
        const float sc = lds[OFF_S2 + c];
        const float sh = lds[OFF_T2 + c];

        for (int p = tid >> 7; p < NPIX_T; p += 2) {
            const int ty = p / TILE;
            const int tx = p - ty * TILE;
            float acc = 0.f;
            #pragma unroll
            for (int dy = 0; dy < 3; ++dy)
                #pragma unroll
                for (int dx = 0; dx < 3; ++dx) {
                    const v4f hv = *(const v4f*)&lds[OFF_H1 +
                        ((ty + dy) * HALO + (tx + dx)) * C1 + g4];  // ds_load_b128
                    const int j = (dy * 3 + dx) * 4;
                    acc += hv.x * wreg[j + 0];
                    acc += hv.y * wreg[j + 1];
                    acc += hv.z * wreg[j + 2];
                    acc += hv.w * wreg[j + 3];
                }
            float vv = acc * sc + sh;
            lds[OFF_H2 + p * C1 + c] = vv > 0.f ? vv : 0.f;
        }
    }
    __syncthreads();

    // ---- Phase 3: conv1x1 (128->256) via WMMA; M = 196 (padded 208) ------
    for (int tt = wave; tt < 13 * 16; tt += 8) {
        const int mt = tt >> 4;                       // 0..12
        const int nt = tt & 15;                       // 0..15
        const int n  = nt * 16 + l16;
        const int pA = mt * 16 + l16;                 // A-row pixel, < 208

        v8f acc = {0.f, 0.f, 0.f, 0.f, 0.f, 0.f, 0.f, 0.f};
        #pragma unroll 8
        for (int ks = 0; ks < C1 / 4; ++ks) {
            const int k = ks * 4 + lhalf * 2;
            const v2f a = *(const v2f*)&lds[OFF_H2 + pA * C1 + k]; // ds_load_b64
            v2f bb;
            bb.x = w3[k * C2 + n];
            bb.y = w3[(k + 1) * C2 + n];
            acc = __builtin_amdgcn_wmma_f32_16x16x4_f32(
                false, a, false, bb, (short)0, acc, false, false);
        }
        const float sc = lds[OFF_S3 + n];
        const float sh = lds[OFF_T3 + n];
        #pragma unroll
        for (int r = 0; r < 8; ++r) {
            const int m = r + lhalf * 8;
            const int p = mt * 16 + m;
            if (p < NPIX_T) {
                const int py = ty0 + p / TILE;
                const int px = tx0 + (p % TILE);
                float vv = acc[r] * sc + sh;
                vv = vv > 0.f ? vv : 0.f;
                out[(((size_t)b * HWDIM + py) * HWDIM + px) * C2 + n] = vv;
            }
        }
    }
}

extern "C" void kernel_launch(void* const* d_in, const int* in_sizes, int n_in,
                              void* d_out, int out_size, void* d_ws, size_t ws_size,
                              hipStream_t stream) {
    (void)in_sizes; (void)n_in; (void)out_size; (void)d_ws; (void)ws_size;
    const float* x   = (const float*)d_in[0];
    const float* w1  = (const float*)d_in[1];
    const float* b1  = (const float*)d_in[2];
    const float* g1  = (const float*)d_in[3];
    const float* be1 = (const float*)d_in[4];
    const float* m1  = (const float*)d_in[5];
    const float* v1  = (const float*)d_in[6];
    const float* wg  = (const float*)d_in[7];
    const float* bg  = (const float*)d_in[8];
    const float* g2  = (const float*)d_in[9];
    const float* be2 = (const float*)d_in[10];
    const float* m2  = (const float*)d_in[11];
    const float* v2  = (const float*)d_in[12];
    const float* w3  = (const float*)d_in[13];
    const float* b3  = (const float*)d_in[14];
    const float* g3  = (const float*)d_in[15];
    const float* be3 = (const float*)d_in[16];
    const float* m3  = (const float*)d_in[17];
    const float* v3  = (const float*)d_in[18];
    float* outp = (float*)d_out;

    // Allow large dynamic LDS (CDNA5: up to 320KB per workgroup). Ignore
    // failure; deterministic and capture-safe (host-side attribute set).
    (void)hipFuncSetAttribute((const void*)resnext_fused_kernel,
                              hipFuncAttributeMaxDynamicSharedMemorySize,
                              LDS_BYTES);

    dim3 grid(BATCH * 16);   // 32 images x (4x4 tiles of 14x14)
    dim3 block(256);         // 8 wave32 waves
    resnext_fused_kernel<<<grid, block, LDS_BYTES, stream>>>(
        x, w1, b1, g1, be1, m1, v1, wg, bg, g2, be2, m2, v2,
        w3, b3, g3, be3, m3, v3, outp);
}